// SimpleMemoryNetwork_2024_84765474554643
// MI455X (gfx1250) — compile-verified
//
#include <hip/hip_runtime.h>
#include <cmath>

// Problem constants (fixed by the reference).
#define B_       128
#define H_       256
#define NO_      10
#define T_       64
#define LAMBDA_  0.95f
#define ETA_     0.5f
#define SAMPLE_  32
#define DECISION_ 8
#define RANK_    (SAMPLE_ + 1)   // 33: eta==0 for t>32, so rank(A) <= 33

typedef float v2f __attribute__((ext_vector_type(2)));
typedef float v8f __attribute__((ext_vector_type(8)));

// Per-step fast-weight coefficients c[s] = ETA * LAMBDA^(t-1-s), computed on
// host, passed by value (132 B of kernarg) -> deterministic, no extra state.
struct Coef {
  int   S;            // number of valid history entries (= min(t, 33))
  float c[RANK_];
};

// ---------------------------------------------------------------------------
// One 16x16 output tile of  act[128,256] @ W[256,256]  accumulated via
// V_WMMA_F32_16X16X4_F32 (K-loop of 64 steps of K=4).
// Fragment layout per CDNA5 ISA (wave32):
//   A 16x4 : lanes 0-15 -> rows, VGPR0=K0,VGPR1=K1; lanes 16-31 -> K2,K3
//   B 4x16 : lanes 0-15 -> cols, VGPR0=K0,VGPR1=K1; lanes 16-31 -> K2,K3
//   C/D    : VGPR v, lanes 0-15 -> (M=v, N=lane), lanes 16-31 -> (M=v+8, ...)
// ---------------------------------------------------------------------------
__device__ __forceinline__ v8f gemm_tile_acc(const float* __restrict__ act,
                                             const float* __restrict__ W,
                                             int mt, int nt, v8f acc) {
  const int lane = threadIdx.x & 31;
  const int l15  = lane & 15;
  const int hi   = lane >> 4;
  const float* arow = act + (mt * 16 + l15) * H_;
  const float* wcol = W + nt * 16 + l15;
#pragma unroll 4
  for (int k0 = 0; k0 < H_; k0 += 4) {
    const int ka = k0 + 2 * hi;
    v2f af; af.x = arow[ka];      af.y = arow[ka + 1];
    v2f bf; bf.x = wcol[ka * H_]; bf.y = wcol[(ka + 1) * H_];
    acc = __builtin_amdgcn_wmma_f32_16x16x4_f32(false, af, false, bf,
                                                (short)0, acc, false, false);
  }
  return acc;
}

// ---------------------------------------------------------------------------
// K1: recurrent populations (WMMA, 256 tiles, one per wave).
//   a_new = relu(x[:,2:5] @ W_ma + a_old @ W_a + r_old @ W_ra)
//   r_new = relu(x[:,0:2] @ W_sr + a_old @ W_ar)
// ---------------------------------------------------------------------------
__global__ void k1_recurrent(const float* __restrict__ x_in,
                             const float* __restrict__ W_sr,
                             const float* __restrict__ W_ma,
                             const float* __restrict__ W_ar,
                             const float* __restrict__ W_ra,
                             const float* __restrict__ W_a,
                             const float* __restrict__ r_old,
                             const float* __restrict__ a_old,
                             float* __restrict__ r_new,
                             float* __restrict__ a_new,
                             int t) {
  const int warp = threadIdx.x >> 5;
  const int tile = blockIdx.x * (blockDim.x >> 5) + warp;  // wave-uniform
  const int lane = threadIdx.x & 31;
  const int l15  = lane & 15;
  const int hi   = lane >> 4;
  v8f acc = {};
  if (tile < 128) {                       // ---- a_new tile
    const int mt = tile >> 4, nt = tile & 15;
    acc = gemm_tile_acc(a_old, W_a,  mt, nt, acc);
    acc = gemm_tile_acc(r_old, W_ra, mt, nt, acc);
    const int h = nt * 16 + l15;
#pragma unroll
    for (int v = 0; v < 8; ++v) {
      const int b = mt * 16 + v + 8 * hi;
      const float* xb = x_in + ((size_t)t * B_ + b) * 5;
      float val = acc[v]
                + xb[2] * W_ma[0 * H_ + h]
                + xb[3] * W_ma[1 * H_ + h]
                + xb[4] * W_ma[2 * H_ + h];
      a_new[b * H_ + h] = fmaxf(val, 0.0f);
    }
  } else {                                // ---- r_new tile
    const int tr = tile - 128;
    const int mt = tr >> 4, nt = tr & 15;
    acc = gemm_tile_acc(a_old, W_ar, mt, nt, acc);
    const int h = nt * 16 + l15;
#pragma unroll
    for (int v = 0; v < 8; ++v) {
      const int b = mt * 16 + v + 8 * hi;
      const float* xb = x_in + ((size_t)t * B_ + b) * 5;
      float val = acc[v]
                + xb[0] * W_sr[0 * H_ + h]
                + xb[1] * W_sr[1 * H_ + h];
      r_new[b * H_ + h] = fmaxf(val, 0.0f);
    }
  }
}

// ---------------------------------------------------------------------------
// K2a: o_pre = r_new @ W_ro + a_new @ W_ao   (WMMA, 128 tiles)
// ---------------------------------------------------------------------------
__global__ void k2a_readout_gemm(const float* __restrict__ r_new,
                                 const float* __restrict__ a_new,
                                 const float* __restrict__ W_ro,
                                 const float* __restrict__ W_ao,
                                 float* __restrict__ o_pre) {
  const int warp = threadIdx.x >> 5;
  const int tile = blockIdx.x * (blockDim.x >> 5) + warp;
  const int lane = threadIdx.x & 31;
  const int l15  = lane & 15;
  const int hi   = lane >> 4;
  const int mt = tile >> 4, nt = tile & 15;
  v8f acc = {};
  acc = gemm_tile_acc(r_new, W_ro, mt, nt, acc);
  acc = gemm_tile_acc(a_new, W_ao, mt, nt, acc);
  const int h = nt * 16 + l15;
#pragma unroll
  for (int v = 0; v < 8; ++v) {
    const int b = mt * 16 + v + 8 * hi;
    o_pre[b * H_ + h] = acc[v];
  }
}

// ---------------------------------------------------------------------------
// K2b: low-rank fast-weight readout.
//   o_pre[b,k] += sum_s c[s] * (r_t[b]·o_s[b]) * r_s[b,k]
// Phase 1: 8 waves compute the <=33 batched dots with lane-strided partials
//          + __shfl_xor wave reductions (emits ds/permute ops).
// Phase 2: each thread recombines its column k from R-history (coalesced).
// Replaces a 96 MB/step streamed-A pass with <=8.4 MB/step of history reads.
// ---------------------------------------------------------------------------
__global__ void k2b_lowrank(const float* __restrict__ r_new,
                            const float* __restrict__ Ohist, // [RANK,B,H]
                            const float* __restrict__ Rhist, // [RANK,B,H]
                            float* __restrict__ o_pre,
                            Coef cf) {
  __shared__ float r_sh[H_];
  __shared__ float d_sh[RANK_];
  const int b    = blockIdx.x;
  const int k    = threadIdx.x;
  const int lane = threadIdx.x & 31;
  const int warp = threadIdx.x >> 5;

  r_sh[k] = r_new[b * H_ + k];
  __syncthreads();

  // Phase 1: d[s] = r_t[b] · o_s[b]
  for (int s = warp; s < cf.S; s += 8) {
    const float* Ob = Ohist + ((size_t)s * B_ + b) * H_;
    float p = 0.0f;
#pragma unroll
    for (int i = 0; i < 8; ++i)
      p = fmaf(r_sh[lane + 32 * i], Ob[lane + 32 * i], p);
#pragma unroll
    for (int off = 16; off > 0; off >>= 1)
      p += __shfl_xor(p, off, 32);
    if (lane == 0) d_sh[s] = p;
  }
  __syncthreads();

  // Phase 2: recombine along k.
  float m = 0.0f;
  for (int s = 0; s < cf.S; ++s)
    m = fmaf(cf.c[s] * d_sh[s], Rhist[((size_t)s * B_ + b) * H_ + k], m);
  o_pre[b * H_ + k] += m;
}

// ---------------------------------------------------------------------------
// K3: deterministic batch-mean norm + relu, history append, readout y.
// ---------------------------------------------------------------------------
__global__ void k3_norm_update(const float* __restrict__ o_pre,
                               const float* __restrict__ r_new,
                               const float* __restrict__ g,
                               const float* __restrict__ bb,
                               const float* __restrict__ W_y,   // [H,NO]
                               float* __restrict__ Ohist,
                               float* __restrict__ Rhist,
                               float* __restrict__ y_out,       // [DECISION,B,NO]
                               int t) {
  __shared__ float o_sh[H_];
  const int b = blockIdx.x;
  const int k = threadIdx.x;

  float mu = 0.0f;
#pragma unroll 4
  for (int b2 = 0; b2 < B_; ++b2)
    mu += o_pre[b2 * H_ + k];
  mu *= (1.0f / (float)B_);

  float o = g[k] * (o_pre[b * H_ + k] - mu) + bb[k];
  o = fmaxf(o, 0.0f);
  o_sh[k] = o;

  // Append (o_t, r_t) to the fast-weight history while eta != 0.
  if (t <= SAMPLE_) {
    Ohist[((size_t)t * B_ + b) * H_ + k] = o;
    Rhist[((size_t)t * B_ + b) * H_ + k] = r_new[b * H_ + k];
  }
  __syncthreads();

  if (t >= T_ - DECISION_ && k < NO_) {
    float acc = 0.0f;
    for (int h = 0; h < H_; ++h)
      acc = fmaf(o_sh[h], W_y[h * NO_ + k], acc);
    y_out[(((size_t)(t - (T_ - DECISION_))) * B_ + b) * NO_ + k] = acc;
  }
}

// ---------------------------------------------------------------------------
// K0: zero r0/a0 each call so graph replays are deterministic.
// (History slots are always written at step s before being read at s+1.)
// ---------------------------------------------------------------------------
__global__ void k0_init(float* __restrict__ p, int n) {
  const int i = blockIdx.x * blockDim.x + threadIdx.x;
  if (i < n) p[i] = 0.0f;
}

extern "C" void kernel_launch(void* const* d_in, const int* in_sizes, int n_in,
                              void* d_out, int out_size, void* d_ws, size_t ws_size,
                              hipStream_t stream) {
  (void)in_sizes; (void)n_in; (void)out_size; (void)ws_size;
  const float* x_in = (const float*)d_in[0];
  const float* W_sr = (const float*)d_in[1];
  const float* W_ma = (const float*)d_in[2];
  const float* W_ro = (const float*)d_in[3];
  const float* W_ao = (const float*)d_in[4];
  const float* W_ar = (const float*)d_in[5];
  const float* W_ra = (const float*)d_in[6];
  const float* W_a  = (const float*)d_in[7];
  const float* W_y  = (const float*)d_in[8];
  const float* g    = (const float*)d_in[9];
  const float* bb   = (const float*)d_in[10];
  float* y_out = (float*)d_out;
  float* ws = (float*)d_ws;

  // Workspace (floats): r0|r1|a0|a1|o_pre | Ohist | Rhist   (~9.5 MB total)
  const size_t BH = (size_t)B_ * H_;
  float* r0    = ws;
  float* r1    = r0 + BH;
  float* a0    = r1 + BH;
  float* a1    = a0 + BH;
  float* o_pre = a1 + BH;
  float* Ohist = o_pre + BH;
  float* Rhist = Ohist + (size_t)RANK_ * BH;

  // Zero initial recurrent state (2*BH floats, contiguous: r0 then r1..a0?
  // r0 and a0 are not adjacent, zero the four state buffers together).
  k0_init<<<(unsigned)((4 * BH + 255) / 256), 256, 0, stream>>>(r0, (int)(4 * BH));

  for (int t = 0; t < T_; ++t) {
    const float* r_old = (t & 1) ? r1 : r0;
    const float* a_old = (t & 1) ? a1 : a0;
    float* r_new = (t & 1) ? r0 : r1;
    float* a_new = (t & 1) ? a0 : a1;

    // Host-computed fast-weight coefficients for this step.
    Coef cf;
    cf.S = (t < RANK_) ? t : RANK_;
    for (int s = 0; s < cf.S; ++s)
      cf.c[s] = ETA_ * powf(LAMBDA_, (float)(t - 1 - s));
    for (int s = cf.S; s < RANK_; ++s) cf.c[s] = 0.0f;

    k1_recurrent<<<32, 256, 0, stream>>>(x_in, W_sr, W_ma, W_ar, W_ra, W_a,
                                         r_old, a_old, r_new, a_new, t);
    k2a_readout_gemm<<<16, 256, 0, stream>>>(r_new, a_new, W_ro, W_ao, o_pre);
    k2b_lowrank<<<B_, H_, 0, stream>>>(r_new, Ohist, Rhist, o_pre, cf);
    k3_norm_update<<<B_, H_, 0, stream>>>(o_pre, r_new, g, bb, W_y,
                                          Ohist, Rhist, y_out, t);
  }
}